// CGAMotorModel_61151744360845
// MI455X (gfx1250) — compile-verified
//
#include <hip/hip_runtime.h>

// ---------------------------------------------------------------------------
// CGA motor model, collapsed linear form.
//   pred[b,k] = sum_{d,p1} x[b,d,p1] * M[(d,p1),k]
//   M[(d,p1),k] = s * sum_p csign(p1,p1^p)*csign(p,p^k) * G[d][p1^p][p^k]
//   G[d][q1][q2] = sum_h W_in[h,d,q1] * W_out[h,q2]
//   s = 1 - 0.9^10   (closed form of the 10-step EP relaxation from h0=0)
// Stages 1 and 3 are f32 WMMA GEMMs (V_WMMA_F32_16X16X4_F32).
// ---------------------------------------------------------------------------

typedef __attribute__((ext_vector_type(2))) float v2f;
typedef __attribute__((ext_vector_type(8))) float v8f;

#define SCALE 0.6513215599f  // 1 - 0.9^10

// Cayley sign for e_a * e_b = sign * e_{a^b}, Cl(4,1) metric diag(1,1,1,1,-1)
__device__ __forceinline__ float csign(int a, int b) {
    int cnt = 0;
    for (int aa = a >> 1; aa; aa >>= 1) cnt += __popc(aa & b);
    if ((a & b) & 16) cnt ^= 1;   // e5^2 = -1 (bit 4)
    return (cnt & 1) ? -1.0f : 1.0f;
}

// ---------------------------------------------------------------------------
// Stage 1: G[d] (32x32) = W_in[:,d,:]^T (32x1024) @ W_out (1024x32)
// One block per d; 8 waves each own a K-chunk of 128, all 4 16x16 tiles,
// partials reduced through LDS.
// ---------------------------------------------------------------------------
__global__ __launch_bounds__(256) void gram_kernel(const float* __restrict__ Win,
                                                   const float* __restrict__ Wout,
                                                   float* __restrict__ G) {
    const int d    = blockIdx.x;      // 0..4
    const int tid  = threadIdx.x;
    const int wid  = tid >> 5;        // wave 0..7
    const int lane = tid & 31;
    const int half = lane & 15;
    const int hi   = lane >> 4;       // 0: K pair {0,1}, 1: K pair {2,3}

    __shared__ float red[8 * 1024];   // 8 partial 32x32 tiles

    v8f acc00 = {}, acc01 = {}, acc10 = {}, acc11 = {};

    const float* WinD = Win + d * 32; // A[m][k] = Win[k*160 + d*32 + m]
    const int kbeg = wid * 128;
    #pragma unroll 4
    for (int k0 = kbeg; k0 < kbeg + 128; k0 += 4) {
        const int kA = k0 + hi * 2;
        v2f a0, a1, b0, b1;
        // A fragments: rows m = q1 (tile 0: m 0..15, tile 1: m 16..31)
        a0.x = WinD[(size_t)kA * 160 + half];
        a0.y = WinD[(size_t)(kA + 1) * 160 + half];
        a1.x = WinD[(size_t)kA * 160 + 16 + half];
        a1.y = WinD[(size_t)(kA + 1) * 160 + 16 + half];
        // B fragments: B[k][n] = Wout[k*32 + n]
        b0.x = Wout[kA * 32 + half];
        b0.y = Wout[(kA + 1) * 32 + half];
        b1.x = Wout[kA * 32 + 16 + half];
        b1.y = Wout[(kA + 1) * 32 + 16 + half];

        acc00 = __builtin_amdgcn_wmma_f32_16x16x4_f32(false, a0, false, b0, (short)0, acc00, false, false);
        acc01 = __builtin_amdgcn_wmma_f32_16x16x4_f32(false, a0, false, b1, (short)0, acc01, false, false);
        acc10 = __builtin_amdgcn_wmma_f32_16x16x4_f32(false, a1, false, b0, (short)0, acc10, false, false);
        acc11 = __builtin_amdgcn_wmma_f32_16x16x4_f32(false, a1, false, b1, (short)0, acc11, false, false);
    }

    // Spill this wave's 32x32 partial: C/D layout m = r + 8*hi, n = half
    float* my = red + wid * 1024;
    #pragma unroll
    for (int r = 0; r < 8; ++r) {
        const int m = r + 8 * hi;
        my[m * 32 + half]             = acc00[r];
        my[m * 32 + 16 + half]        = acc01[r];
        my[(m + 16) * 32 + half]      = acc10[r];
        my[(m + 16) * 32 + 16 + half] = acc11[r];
    }
    __syncthreads();

    // Cross-wave reduce: 1024 outputs, 256 threads x 4
    #pragma unroll
    for (int j = 0; j < 4; ++j) {
        const int o = tid * 4 + j;
        float s = 0.0f;
        #pragma unroll
        for (int w = 0; w < 8; ++w) s += red[w * 1024 + o];
        G[d * 1024 + o] = s;
    }
}

// ---------------------------------------------------------------------------
// Stage 2: sign-mix G -> M (160x32). Each block stages its G[d] in LDS.
// ---------------------------------------------------------------------------
__global__ __launch_bounds__(256) void mix_kernel(const float* __restrict__ G,
                                                  float* __restrict__ M) {
    __shared__ float Gd[1024];
    const int t = blockIdx.x * 256 + threadIdx.x;  // 0..5119, d constant per block
    const int d = t >> 10;
    const float* src = G + d * 1024;
    for (int j = threadIdx.x; j < 1024; j += 256) Gd[j] = src[j];
    __syncthreads();

    const int p1 = (t >> 5) & 31;
    const int k  = t & 31;
    float acc = 0.0f;
    #pragma unroll
    for (int p = 0; p < 32; ++p) {
        const int q1 = p1 ^ p;
        const int q2 = p ^ k;
        acc += csign(p1, q1) * csign(p, q2) * Gd[q1 * 32 + q2];
    }
    M[((d << 5) | p1) * 32 + k] = SCALE * acc;
}

// ---------------------------------------------------------------------------
// Stage 3: pred (512x32) = x (512x160) @ M (160x32).
// 64 waves: 32 row tiles x 2 col tiles, K loop of 40 WMMAs.
// ---------------------------------------------------------------------------
__global__ __launch_bounds__(256) void apply_kernel(const float* __restrict__ X,
                                                    const float* __restrict__ M,
                                                    float* __restrict__ out) {
    const int tid  = threadIdx.x;
    const int wid  = tid >> 5;
    const int lane = tid & 31;
    const int half = lane & 15;
    const int hi   = lane >> 4;
    const int tile = blockIdx.x * 8 + wid;  // 0..63
    const int rt   = tile >> 1;             // row tile 0..31
    const int ct   = tile & 1;              // col tile 0..1

    v8f acc = {};
    const float* Xrow = X + (size_t)rt * 16 * 160;
    #pragma unroll 8
    for (int k0 = 0; k0 < 160; k0 += 4) {
        const int kA = k0 + hi * 2;
        v2f a, b;
        a.x = Xrow[(size_t)half * 160 + kA];
        a.y = Xrow[(size_t)half * 160 + kA + 1];
        b.x = M[kA * 32 + ct * 16 + half];
        b.y = M[(kA + 1) * 32 + ct * 16 + half];
        acc = __builtin_amdgcn_wmma_f32_16x16x4_f32(false, a, false, b, (short)0, acc, false, false);
    }
    #pragma unroll
    for (int r = 0; r < 8; ++r) {
        const int m = rt * 16 + r + 8 * hi;
        out[m * 32 + ct * 16 + half] = acc[r];
    }
}

extern "C" void kernel_launch(void* const* d_in, const int* in_sizes, int n_in,
                              void* d_out, int out_size, void* d_ws, size_t ws_size,
                              hipStream_t stream) {
    const float* x    = (const float*)d_in[0];  // (512, 5, 32)
    const float* Win  = (const float*)d_in[1];  // (1024, 5, 32)
    const float* Wout = (const float*)d_in[2];  // (1, 1024, 32)
    float* out = (float*)d_out;                 // (512, 1, 32)

    float* G = (float*)d_ws;                    // 5*32*32 floats
    float* M = G + 5 * 1024;                    // 160*32  floats

    gram_kernel<<<5, 256, 0, stream>>>(Win, Wout, G);
    mix_kernel<<<20, 256, 0, stream>>>(G, M);
    apply_kernel<<<8, 256, 0, stream>>>(x, M, out);
}